// TrajectoryScore_36481452212940
// MI455X (gfx1250) — compile-verified
//
#include <hip/hip_runtime.h>
#include <stdint.h>

// ---------------- problem geometry ----------------
#define BATCH              64
#define FLOATS_PER_BATCH   (1000 * 128 * 3)        // 384000 floats, contiguous per batch
#define THREADS            256                      // 8 wave32 per block
#define WAVES              (THREADS / 32)
#define F4_PER_THREAD      3                        // 48 B / thread / stage -> 4 triples
#define F4_PER_STAGE       (THREADS * F4_PER_THREAD)   // 768 float4 = 3072 floats = 12 KB
#define STAGES_PER_BATCH   (FLOATS_PER_BATCH / (F4_PER_STAGE * 4))  // 125
#define CHUNKS             25
#define STAGES_PER_BLOCK   (STAGES_PER_BATCH / CHUNKS) // 5
// THRESH2 = (2*sin(deg2rad(2)/2))^2 = 2*(1-cos(2 deg))
#define THRESH2            1.21834596e-03f

typedef __attribute__((ext_vector_type(2))) float v2f;
typedef __attribute__((ext_vector_type(8))) float v8f;

// ---------------- CDNA5 async-to-LDS helpers ----------------
// GLOBAL_LOAD_ASYNC_TO_LDS_B128 (VGLOBAL op 98): vdst = LDS byte offset VGPR,
// vaddr = 64-bit global address VGPR pair, no saddr. Tracked by ASYNCcnt.
__device__ __forceinline__ void async_ld_b128(uint32_t lds_byte_off, const void* gptr) {
    asm volatile("global_load_async_to_lds_b128 %0, %1, off"
                 :: "v"(lds_byte_off), "v"((uint64_t)(uintptr_t)gptr)
                 : "memory");
}
__device__ __forceinline__ void wait_async_le3() {
    asm volatile("s_wait_asynccnt 3" ::: "memory");
}
__device__ __forceinline__ void wait_async_0() {
    asm volatile("s_wait_asynccnt 0" ::: "memory");
}

// ---------------- streaming reduction kernel ----------------
__global__ __launch_bounds__(THREADS)
void trajscore_partials(const float* __restrict__ z,
                        const float* __restrict__ R,
                        float* __restrict__ partials) {
    __shared__ float4 sbuf[2][F4_PER_STAGE];   // 2 x 12 KB double buffer
    __shared__ float  wred[WAVES];

    const int tid   = threadIdx.x;
    const int chunk = blockIdx.x;   // 0..24
    const int b     = blockIdx.y;   // 0..63

    const float r  = R[b];
    const float Bc = -0.5f / (r * r);

    // this thread's first float4 of its chunk (global)
    const float4* gbase = reinterpret_cast<const float4*>(z)
                        + (size_t)b * (FLOATS_PER_BATCH / 4)
                        + (size_t)chunk * STAGES_PER_BLOCK * F4_PER_STAGE
                        + (size_t)tid * F4_PER_THREAD;

    // LDS byte offsets of this thread's slice in each buffer
    // (flat addr of __shared__ truncated to 32 bits == workgroup-relative LDS offset)
    const uint32_t lds0 = (uint32_t)(uintptr_t)&sbuf[0][tid * F4_PER_THREAD];
    const uint32_t lds1 = (uint32_t)(uintptr_t)&sbuf[1][tid * F4_PER_THREAD];

    // prologue: stage 0 -> buf 0   (3 async ops per wave -> ASYNCcnt = 3)
    async_ld_b128(lds0 +  0, gbase + 0);
    async_ld_b128(lds0 + 16, gbase + 1);
    async_ld_b128(lds0 + 32, gbase + 2);

    float acc = 0.0f;
    #pragma unroll
    for (int s = 0; s < STAGES_PER_BLOCK; ++s) {
        if (s + 1 < STAGES_PER_BLOCK) {
            // prefetch next stage into the other buffer, then wait for stage s
            const float4* g = gbase + (size_t)(s + 1) * F4_PER_STAGE;
            const uint32_t l = ((s + 1) & 1) ? lds1 : lds0;
            async_ld_b128(l +  0, g + 0);
            async_ld_b128(l + 16, g + 1);
            async_ld_b128(l + 32, g + 2);
            wait_async_le3();   // stage s's 3 ops retired; stage s+1 still in flight
        } else {
            wait_async_0();     // drain
        }

        // each lane consumes exactly the slice it requested -> no block barrier needed
        const float4* s4 = &sbuf[s & 1][tid * F4_PER_THREAD];
        const float4 a = s4[0];
        const float4 c = s4[1];
        const float4 d = s4[2];

        const float z2_0 = a.x*a.x + a.y*a.y + a.z*a.z;
        const float z2_1 = a.w*a.w + c.x*c.x + c.y*c.y;
        const float z2_2 = c.z*c.z + c.w*c.w + d.x*d.x;
        const float z2_3 = d.y*d.y + d.z*d.z + d.w*d.w;

        // ~1.2e-3 hit rate: branch so whole waves usually skip v_exp_f32
        if (z2_0 < THRESH2) acc += __expf(Bc * z2_0);
        if (z2_1 < THRESH2) acc += __expf(Bc * z2_1);
        if (z2_2 < THRESH2) acc += __expf(Bc * z2_2);
        if (z2_3 < THRESH2) acc += __expf(Bc * z2_3);
    }

    // ---- wave reduction via the matrix pipe ----
    // A (16x4 f32, 2 VGPRs/lane): A[m,0]=acc_m, A[m,2]=acc_{m+16}, K=1,3 zero.
    // B = ones  =>  D[m][n] = acc_m + acc_{m+16}.  Sum the 8 D VGPRs per lane,
    // then one cross-half shuffle gives the full 32-lane sum in every lane.
    const int lane = tid & 31;
    const int wave = tid >> 5;
    float wavesum;
#if __has_builtin(__builtin_amdgcn_wmma_f32_16x16x4_f32)
    {
        v2f a;    a[0] = acc;  a[1] = 0.0f;
        v2f ones; ones[0] = 1.0f; ones[1] = 1.0f;
        v8f cz = {};
        v8f dd = __builtin_amdgcn_wmma_f32_16x16x4_f32(
                     false, a, false, ones, (short)0, cz, false, false);
        float h = ((dd[0] + dd[1]) + (dd[2] + dd[3]))
                + ((dd[4] + dd[5]) + (dd[6] + dd[7]));
        wavesum = h + __shfl_xor(h, 16, 32);
    }
#else
    {
        float v = acc;
        #pragma unroll
        for (int off = 16; off > 0; off >>= 1) v += __shfl_xor(v, off, 32);
        wavesum = v;
    }
#endif

    if (lane == 0) wred[wave] = wavesum;
    __syncthreads();
    if (tid == 0) {
        float t = 0.0f;
        #pragma unroll
        for (int w = 0; w < WAVES; ++w) t += wred[w];
        partials[b * CHUNKS + chunk] = t;
    }
}

// ---------------- finalize: sum partials + analytic tail ----------------
__global__ __launch_bounds__(64)
void trajscore_finalize(const float* __restrict__ partials,
                        const float* __restrict__ R,
                        const float* __restrict__ num_obs_p,
                        float* __restrict__ out) {
    const int b = threadIdx.x;
    if (b >= BATCH) return;

    float raw = 0.0f;
    #pragma unroll
    for (int c = 0; c < CHUNKS; ++c) raw += partials[b * CHUNKS + c];

    const float r    = R[b];
    const float A    = 1.0f / (r * r);
    const float half = 0.5f * A * THRESH2;
    const float mu_per  = (1.0f - expf(-half))        / (2.0f * A);
    const float e2      = (1.0f - expf(-2.0f * half)) / (4.0f * A);
    const float var_per = e2 - mu_per * mu_per;
    const float n  = num_obs_p[0];
    const float mu = n * mu_per;
    const float s2 = n * var_per;

    out[          b] = raw;                 // raw_score
    out[BATCH   + b] = mu;                  // mu
    out[2*BATCH + b] = s2;                  // sigma2
    out[3*BATCH + b] = raw - mu - s2;       // objective (ALPHA = BETA = 1)
}

// ---------------- entry point ----------------
extern "C" void kernel_launch(void* const* d_in, const int* in_sizes, int n_in,
                              void* d_out, int out_size, void* d_ws, size_t ws_size,
                              hipStream_t stream) {
    const float* z       = (const float*)d_in[0];   // (64,1000,128,3) f32
    const float* R       = (const float*)d_in[1];   // (64,1) f32
    const float* num_obs = (const float*)d_in[2];   // scalar f32
    float* out      = (float*)d_out;                // 256 floats
    float* partials = (float*)d_ws;                 // 64*25 floats, fully rewritten each call

    dim3 grid(CHUNKS, BATCH);
    trajscore_partials<<<grid, THREADS, 0, stream>>>(z, R, partials);
    trajscore_finalize<<<1, 64, 0, stream>>>(partials, R, num_obs, out);
}